// ProtoConLoss_12927851560992
// MI455X (gfx1250) — compile-verified
//
#include <hip/hip_runtime.h>
#include <math.h>

#define N_SAMPLES 8192
#define DIM       256
#define NUM_CLS   100
#define CPAD      128          // classes padded to 8 column tiles of 16
#define EPSN      1e-8f

typedef __attribute__((ext_vector_type(2))) float v2f;
typedef __attribute__((ext_vector_type(8))) float v8f;

// -------- workspace layout (floats) --------
// [0,      32768)  protoPair[k2][c][2] : element ((k>>1)*CPAD + c)*2 + (k&1)
//                  = proto[c][k]; zero-padded for c in [100,128)
// [32768,  32896)  inv_pnorm[128]
// [32896,  33024)  countf[128]
// [33024,  41216)  loss_row[8192]

// ---- 1) class prototypes: one block per (padded) class ----
__global__ __launch_bounds__(256)
void proto_kernel(const float* __restrict__ F, const int* __restrict__ labels,
                  float* __restrict__ protoPair, float* __restrict__ inv_pnorm,
                  float* __restrict__ countf)
{
    const int c   = blockIdx.x;      // 0..127
    const int tid = threadIdx.x;     // dim index k = 0..255
    float sum = 0.0f;
    int   cnt = 0;
    if (c < NUM_CLS) {
        for (int i = 0; i < N_SAMPLES; ++i) {
            if (labels[i] == c) { sum += F[(size_t)i * DIM + tid]; ++cnt; }
        }
    }
    const float cf = (float)cnt;
    const float pd = sum / fmaxf(cf, 1.0f);                   // mean prototype element
    protoPair[(((size_t)(tid >> 1)) * CPAD + c) * 2 + (tid & 1)] = pd;  // K-pair layout

    __shared__ float red[256];
    red[tid] = pd * pd;
    __syncthreads();
    for (int s = 128; s > 0; s >>= 1) {
        if (tid < s) red[tid] += red[tid + s];
        __syncthreads();
    }
    if (tid == 0) {
        const float pn = fmaxf(sqrtf(red[0]), EPSN);
        inv_pnorm[c] = 1.0f / pn;
        countf[c]    = cf;                                    // 0 for padded classes
    }
}

// ---- 2) WMMA GEMM (F @ protoT) fused with f-norms and contrastive epilogue ----
__global__ __launch_bounds__(128)
void gemm_loss_kernel(const float* __restrict__ F, const int* __restrict__ labels,
                      const float* __restrict__ protoPair,
                      const float* __restrict__ inv_pnorm,
                      const float* __restrict__ countf,
                      float* __restrict__ loss_row)
{
    const int wave = threadIdx.x >> 5;
    const int lane = threadIdx.x & 31;
    const int half = lane >> 4;      // 0: lanes 0-15, 1: lanes 16-31
    const int lm   = lane & 15;
    const int row_base = (blockIdx.x * 4 + wave) * 16;

    const v8f vzero = {};
    v8f acc[8];
#pragma unroll
    for (int t = 0; t < 8; ++t) acc[t] = vzero;

    // A fragment: lane lm holds row (row_base+lm); half selects K pair (k,k+1)/(k+2,k+3)
    const float* Arow = F + (size_t)(row_base + lm) * DIM;

    float nacc = 0.0f;               // this lane's half of ||f_{row_base+lm}||^2
    for (int k = 0; k < DIM; k += 4) {
        const int ka = k + 2 * half;
        const v2f a = *(const v2f*)(Arow + ka);               // b64 load (K=ka, ka+1)
        nacc += a.x * a.x + a.y * a.y;
        const float* Bp = protoPair + ((size_t)(ka >> 1) * CPAD) * 2;
#pragma unroll
        for (int t = 0; t < 8; ++t) {
            const v2f b = *(const v2f*)(Bp + 2 * (t * 16 + lm));  // b64 load (K=ka, ka+1)
            acc[t] = __builtin_amdgcn_wmma_f32_16x16x4_f32(
                /*neg_a=*/false, a, /*neg_b=*/false, b,
                /*c_mod=*/(short)0, acc[t], /*reuse_a=*/false, /*reuse_b=*/false);
        }
    }

    // combine the two K-halves: every lane now has full ||f_{row_base+lm}||^2
    nacc += __shfl_xor(nacc, 16, 32);
    const float invT   = 1.0f / 0.7f;
    const float invf_l = invT / fmaxf(sqrtf(nacc), EPSN);     // (1/T)/||f|| for row lm

    // ---- epilogue: acc[t] vgpr r = element (M = r + 8*half, N = t*16 + lm) ----
    float ifn[8];
    int   lab[8];
#pragma unroll
    for (int r = 0; r < 8; ++r) {
        const int row = row_base + 8 * half + r;
        ifn[r] = __shfl(invf_l, 8 * half + r, 32);            // norm lives in lane (row%16)
        lab[r] = labels[row];
    }
    float rs[8], dg[8];
#pragma unroll
    for (int r = 0; r < 8; ++r) { rs[r] = 0.0f; dg[r] = 0.0f; }

#pragma unroll
    for (int t = 0; t < 8; ++t) {
        const int   c  = t * 16 + lm;
        const float cf = countf[c];       // 0 for padded classes -> no contribution
        const float ip = inv_pnorm[c];
#pragma unroll
        for (int r = 0; r < 8; ++r) {
            const float sc = acc[t][r] * ip * ifn[r];   // sim / T
            rs[r] += cf * __expf(sc);
            if (lab[r] == c) dg[r] = sc;
        }
    }

    // reduce across the 16 lanes of this half-wave (xor masks stay within half)
#pragma unroll
    for (int r = 0; r < 8; ++r) {
#pragma unroll
        for (int m = 8; m >= 1; m >>= 1) {
            rs[r] += __shfl_xor(rs[r], m, 32);
            dg[r] += __shfl_xor(dg[r], m, 32);
        }
    }
    if (lm == 0) {
#pragma unroll
        for (int r = 0; r < 8; ++r) {
            const int row = row_base + 8 * half + r;
            loss_row[row] = __logf(rs[r]) - dg[r];
        }
    }
}

// ---- 3) deterministic final reduction ----
__global__ __launch_bounds__(256)
void reduce_kernel(const float* __restrict__ loss_row, float* __restrict__ out)
{
    __shared__ float red[256];
    const int tid = threadIdx.x;
    float s = 0.0f;
    for (int i = tid; i < N_SAMPLES; i += 256) s += loss_row[i];
    red[tid] = s;
    __syncthreads();
    for (int st = 128; st > 0; st >>= 1) {
        if (tid < st) red[tid] += red[tid + st];
        __syncthreads();
    }
    if (tid == 0) out[0] = red[0] / (float)N_SAMPLES;
}

extern "C" void kernel_launch(void* const* d_in, const int* in_sizes, int n_in,
                              void* d_out, int out_size, void* d_ws, size_t ws_size,
                              hipStream_t stream)
{
    const float* F      = (const float*)d_in[0];
    const int*   labels = (const int*)d_in[1];

    float* ws        = (float*)d_ws;
    float* protoPair = ws;             // 32768 floats
    float* inv_pnorm = ws + 32768;     // 128
    float* countf    = ws + 32896;     // 128
    float* loss_row  = ws + 33024;     // 8192
    float* out       = (float*)d_out;

    proto_kernel<<<CPAD, 256, 0, stream>>>(F, labels, protoPair, inv_pnorm, countf);
    gemm_loss_kernel<<<N_SAMPLES / 64, 128, 0, stream>>>(F, labels, protoPair, inv_pnorm,
                                                         countf, loss_row);
    reduce_kernel<<<1, 256, 0, stream>>>(loss_row, out);
}